// FlexiSoftmaxClassifier_15814069583963
// MI455X (gfx1250) — compile-verified
//
#include <hip/hip_runtime.h>

// FlexiSoftmaxClassifier for MI455X (gfx1250, wave32).
//
// Roofline: gather = 512 MB moved (~22 us @ 23.3 TB/s), penalty = 256 MB
// streamed (~11 us). Zero useful FLOPs -> optimize purely for b128 data
// movement; route the penalty summation through V_WMMA_F32_16X16X4_F32
// (all-ones B) so the cross-element adds run on the matrix pipe in f32,
// deterministically, hidden behind the HBM stream.

typedef __attribute__((ext_vector_type(2))) float v2f;
typedef __attribute__((ext_vector_type(4))) float v4f;
typedef __attribute__((ext_vector_type(8))) float v8f;

#define NDIM 8192
#define NBITS 13            // log2(NDIM)
#define PEN_BLOCKS 2048
#define PEN_THREADS 256     // 8 waves (wave32)

// ---------------------------------------------------------------------------
// Kernel 1: laff[b, :] = R[l[b], :]   (one block per output row)
// b128 loads (rows repeat -> let L2 keep them), non-temporal b128 stores
// (write-once 256 MB stream; keep it out of the 192 MB L2).
// ---------------------------------------------------------------------------
__global__ void gather_rows_kernel(const float* __restrict__ R,
                                   const int* __restrict__ l,
                                   float* __restrict__ out) {
  const int b = blockIdx.x;
  const int row = l[b];
  const v4f* __restrict__ src = (const v4f*)(R + (size_t)row * NDIM);
  v4f* __restrict__ dst = (v4f*)(out + (size_t)b * NDIM);
  // NDIM/4 = 2048 float4 per row; 256 threads -> 8 b128 per thread, coalesced.
  #pragma unroll
  for (int i = 0; i < (NDIM / 4) / PEN_THREADS; ++i) {
    const int idx = threadIdx.x + i * PEN_THREADS;
    v4f v = src[idx];
    __builtin_nontemporal_store(v, &dst[idx]);
  }
}

// ---------------------------------------------------------------------------
// Kernel 2: per-block partial sums of (I - R)^2 via WMMA f32 16x16x4.
// Each wave owns chunks of 64 contiguous elements. Lane L loads elements
// {base+2L, base+2L+1} (coalesced b64), squares the residuals, and packs
// them into the two A-operand VGPRs. B = all ones => D[i][j] = sum_k A[i][k],
// accumulated in C across the loop. Final extraction: per-lane sum of the 8
// C VGPRs covers rows 0-7 (lanes 0-15) or rows 8-15 (lanes 16-31); adding
// the lane^16 partner gives the full tile sum in every lane.
// ---------------------------------------------------------------------------
__global__ void penalty_partial_kernel(const float* __restrict__ R,
                                       float* __restrict__ ws) {
  const int lane = threadIdx.x & 31;
  const int waveInBlock = threadIdx.x >> 5;
  const int wavesPerBlock = PEN_THREADS >> 5;                   // 8
  const int totalWaves = PEN_BLOCKS * wavesPerBlock;            // 16384
  const int w = blockIdx.x * wavesPerBlock + waveInBlock;
  const int totalChunks = (NDIM * NDIM) >> 6;                   // 2^20 chunks

  v8f c = {0.f, 0.f, 0.f, 0.f, 0.f, 0.f, 0.f, 0.f};
  v2f bones; bones.x = 1.0f; bones.y = 1.0f;

  // 2^20 / 16384 = 64 iterations per wave; bound is wave-uniform so EXEC
  // stays all-1s (WMMA requirement).
  for (int chunk = w; chunk < totalChunks; chunk += totalWaves) {
    const int e0 = (chunk << 6) + (lane << 1);     // element index, < 2^26
    v2f r = __builtin_nontemporal_load((const v2f*)(R + e0));   // streamed once

    const int i0 = e0 >> NBITS, j0 = e0 & (NDIM - 1);
    const int e1 = e0 + 1;
    const int i1 = e1 >> NBITS, j1 = e1 & (NDIM - 1);
    const float d0 = (i0 == j0) ? 1.0f : 0.0f;     // v_cndmask, no divergence
    const float d1 = (i1 == j1) ? 1.0f : 0.0f;
    const float v0 = d0 - r.x;
    const float v1 = d1 - r.y;
    v2f a; a.x = v0 * v0; a.y = v1 * v1;

    // D = A x ones + C : every column of D holds the per-row sums of A.
    c = __builtin_amdgcn_wmma_f32_16x16x4_f32(
        /*neg_a=*/false, a, /*neg_b=*/false, bones,
        /*c_mod=*/(short)0, c, /*reuse_a=*/false, /*reuse_b=*/false);
  }

  float s = c[0] + c[1] + c[2] + c[3] + c[4] + c[5] + c[6] + c[7];
  s += __shfl_xor(s, 16, 32);   // rows 0-7 + rows 8-15 -> full tile sum

  __shared__ float sm[8];
  if (lane == 0) sm[waveInBlock] = s;
  __syncthreads();
  if (threadIdx.x == 0) {
    float t = 0.0f;                      // fixed-order -> deterministic
    #pragma unroll
    for (int i = 0; i < 8; ++i) t += sm[i];
    ws[blockIdx.x] = t;
  }
}

// ---------------------------------------------------------------------------
// Kernel 3: deterministic tree reduction of the 2048 block partials.
// ---------------------------------------------------------------------------
__global__ void penalty_final_kernel(const float* __restrict__ ws,
                                     float* __restrict__ outp) {
  __shared__ float sm[PEN_THREADS];
  float t = 0.0f;
  for (int i = threadIdx.x; i < PEN_BLOCKS; i += PEN_THREADS) t += ws[i];
  sm[threadIdx.x] = t;
  __syncthreads();
  #pragma unroll
  for (int off = PEN_THREADS / 2; off > 0; off >>= 1) {
    if ((int)threadIdx.x < off) sm[threadIdx.x] += sm[threadIdx.x + off];
    __syncthreads();
  }
  if (threadIdx.x == 0) {
    // mean over N^2 = 2^26 elements: exact power-of-two scale.
    outp[0] = sm[0] * (1.0f / 67108864.0f);
  }
}

// ---------------------------------------------------------------------------
extern "C" void kernel_launch(void* const* d_in, const int* in_sizes, int n_in,
                              void* d_out, int out_size, void* d_ws, size_t ws_size,
                              hipStream_t stream) {
  const int*   l = (const int*)d_in[0];      // setup_inputs order: l, R
  const float* R = (const float*)d_in[1];
  float* out = (float*)d_out;                // [B*N] laff, then [1] penalty
  float* pws = (float*)d_ws;                 // PEN_BLOCKS floats (8 KB)

  gather_rows_kernel<<<NDIM, PEN_THREADS, 0, stream>>>(R, l, out);
  penalty_partial_kernel<<<PEN_BLOCKS, PEN_THREADS, 0, stream>>>(R, pws);
  penalty_final_kernel<<<1, PEN_THREADS, 0, stream>>>(
      pws, out + (size_t)NDIM * (size_t)NDIM);
}